// GATGRUCell_2241972929140
// MI455X (gfx1250) — compile-verified
//
#include <hip/hip_runtime.h>
#include <cstddef>
#include <cstdint>

// Problem constants (from reference)
#define NN   50000
#define EE   800000
#define ETOT (EE + NN)     // edges + self loops
#define NEG_SLOPE 0.2f
#define LN_EPS 1e-5f

typedef __attribute__((ext_vector_type(2))) float v2f;
typedef __attribute__((ext_vector_type(8))) float v8f;

struct AllPtrs {
  const float* Wl[6]; const float* bl[6];
  const float* Wr[6]; const float* br[6];
  const float* We[6]; const float* att[6]; const float* b[6];
  const float* ln_g; const float* ln_b;
};

__device__ __forceinline__ float lrelu(float v) { return v > 0.f ? v : NEG_SLOPE * v; }

__device__ __forceinline__ float wave_sum32(float v) {
#pragma unroll
  for (int off = 16; off; off >>= 1) v += __shfl_xor(v, off, 32);
  return v;
}

// monotone-encoding float atomic max (values init to -inf)
__device__ __forceinline__ void atomicMaxF(float* addr, float val) {
  if (val >= 0.f) atomicMax((int*)addr, __float_as_int(val));
  else            atomicMin((unsigned int*)addr, __float_as_uint(val));
}

// ---------------------------------------------------------------------------
// K0: init accumulators (re-run every call: ws is not re-poisoned by harness)
// ---------------------------------------------------------------------------
__global__ void k_init(float* amax, float* denom, float* outacc, float* eaSum) {
  const size_t total = (size_t)6 * NN * 64;
  for (size_t i = blockIdx.x * (size_t)blockDim.x + threadIdx.x; i < total;
       i += (size_t)gridDim.x * blockDim.x) {
    outacc[i] = 0.f;
    if (i < (size_t)NN * 12) { denom[i] = 0.f; amax[i] = -3.402823466e38f; }
    if (i < 16) eaSum[i] = 0.f;
  }
}

// ---------------------------------------------------------------------------
// K1: column sums of edge_attr [E x 16]  (stride keeps k = tid%16 invariant)
// ---------------------------------------------------------------------------
__global__ void k_ea_mean(const float* __restrict__ ea, float* __restrict__ eaSum) {
  __shared__ float s[16];
  if (threadIdx.x < 16) s[threadIdx.x] = 0.f;
  __syncthreads();
  const int k = threadIdx.x & 15;
  float local = 0.f;
  const size_t total = (size_t)EE * 16;
  for (size_t i = blockIdx.x * (size_t)blockDim.x + threadIdx.x; i < total;
       i += (size_t)gridDim.x * blockDim.x)
    local += ea[i];
  atomicAdd(&s[k], local);
  __syncthreads();
  if (threadIdx.x < 16) atomicAdd(&eaSum[threadIdx.x], s[threadIdx.x]);
}

// ---------------------------------------------------------------------------
// K2: epmean[g*64+c] = (mean edge_attr) @ We_g   (self-loop edge projection)
// ---------------------------------------------------------------------------
__global__ void k_prep(AllPtrs P, const float* __restrict__ eaSum, float* __restrict__ epmean) {
  int t = threadIdx.x;
  if (t < 384) {
    int g = t / 64, c = t & 63;
    float s = 0.f;
    const float invE = 1.0f / (float)EE;
#pragma unroll
    for (int k = 0; k < 16; ++k) s += eaSum[k] * invE * P.We[g][k * 64 + c];
    epmean[t] = s;
  }
}

// ---------------------------------------------------------------------------
// K3: pack 6 (Wl,Wr) pairs into [64 x 384] matrices (one per input side) and
//     biases into [384] vectors so the WMMA GEMM streams one weight panel.
//     x-side convs: g=0(xr),2(xu),4(xc)   h-side: g=1(hr),3(hu),5(hc)
// ---------------------------------------------------------------------------
__global__ void k_pack(AllPtrs P, float* Wx, float* bx, float* Wh, float* bh) {
  const int totalW = 64 * 384;
  for (int i = blockIdx.x * blockDim.x + threadIdx.x; i < totalW + 384;
       i += gridDim.x * blockDim.x) {
    if (i < totalW) {
      int k = i / 384, col = i % 384;
      int q = col / 128, side = (col >> 6) & 1, jj = col & 63;
      int gx = 2 * q, gh = 2 * q + 1;
      Wx[i] = side ? P.Wr[gx][k * 64 + jj] : P.Wl[gx][k * 64 + jj];
      Wh[i] = side ? P.Wr[gh][k * 64 + jj] : P.Wl[gh][k * 64 + jj];
    } else {
      int col = i - totalW;
      int q = col / 128, side = (col >> 6) & 1, jj = col & 63;
      int gx = 2 * q, gh = 2 * q + 1;
      bx[col] = side ? P.br[gx][jj] : P.bl[gx][jj];
      bh[col] = side ? P.br[gh][jj] : P.bl[gh][jj];
    }
  }
}

// ---------------------------------------------------------------------------
// K4: WMMA GEMM  Out[N x 384] = X[N x 64] * Wp[64 x 384] + bp
//     One wave per 16x16 output tile; fp32 WMMA (16x16x4) x 16 K-steps.
//     A frag (16x4 f32): lane<16 -> {K,K+1}, lane>=16 -> {K+2,K+3} of row M=lane%16
//     B frag (4x16 f32): same K split, column N = lane%16
//     C/D (16x16 f32):   VGPR i holds row i (lanes 0-15) / row i+8 (lanes 16-31)
// ---------------------------------------------------------------------------
__global__ void k_gemm(const float* __restrict__ X, const float* __restrict__ Wp,
                       const float* __restrict__ bp, float* __restrict__ Out, int nrows) {
  const int lane = threadIdx.x & 31;
  const int m = lane & 15;        // row-in-tile (A) or col-in-tile (B/C/D)
  const int kh = lane >> 4;       // K-half select
  const int rowTiles = (nrows + 15) / 16;
  const int tiles = rowTiles * 24;
  const int wid = (blockIdx.x * blockDim.x + threadIdx.x) >> 5;
  const int nwaves = (gridDim.x * blockDim.x) >> 5;

  for (int t = wid; t < tiles; t += nwaves) {
    const int row0 = (t / 24) * 16;
    const int col0 = (t % 24) * 16;
    int rA = row0 + m; if (rA > nrows - 1) rA = nrows - 1;  // keep EXEC full
    v8f acc = {};
#pragma unroll
    for (int k = 0; k < 64; k += 4) {
      const int kb = k + 2 * kh;
      v2f a, b;
      a.x = X[(size_t)rA * 64 + kb];
      a.y = X[(size_t)rA * 64 + kb + 1];
      b.x = Wp[(size_t)kb * 384 + col0 + m];
      b.y = Wp[(size_t)(kb + 1) * 384 + col0 + m];
      acc = __builtin_amdgcn_wmma_f32_16x16x4_f32(false, a, false, b, (short)0, acc,
                                                  false, false);
    }
    const float bb = bp[col0 + m];
#pragma unroll
    for (int i = 0; i < 8; ++i) {
      const int r = row0 + i + 8 * kh;
      if (r < nrows) Out[(size_t)r * 384 + col0 + m] = acc[i] + bb;
    }
  }
}

// ---------------------------------------------------------------------------
// K5: edge pass A — alpha[e][g][h] + atomic segment-max into amax[dst]
//     one wave per edge; lane L owns channels c0=L (head0) and c1=L+32 (head1)
// ---------------------------------------------------------------------------
__global__ void k_edge_a(const int* __restrict__ ei, const float* __restrict__ ea,
                         const float* __restrict__ outx, const float* __restrict__ outh,
                         const float* __restrict__ epmean, AllPtrs P,
                         float* __restrict__ alphaB, float* __restrict__ amax) {
  __shared__ float WeLds[6 * 1024];   // 6 x [16 x 64]
  __shared__ float attLds[384];       // 6 x 64 (head-major == channel index)
  __shared__ float epmLds[384];
  for (int i = threadIdx.x; i < 6 * 1024; i += blockDim.x) {
    int g = i >> 10; WeLds[i] = P.We[g][i & 1023];
  }
  for (int i = threadIdx.x; i < 384; i += blockDim.x) {
    attLds[i] = P.att[i / 64][i & 63];
    epmLds[i] = epmean[i];
  }
  __syncthreads();

  const int lane = threadIdx.x & 31;
  const int c0 = lane, c1 = lane + 32;
  const int wid = (blockIdx.x * blockDim.x + threadIdx.x) >> 5;
  const int nwaves = (gridDim.x * blockDim.x) >> 5;

  for (int e = wid; e < ETOT; e += nwaves) {
    int src, dst;
    const bool real = e < EE;
    if (real) { src = ei[e]; dst = ei[EE + e]; }
    else      { src = e - EE; dst = src; }

    float eav = real ? ea[(size_t)e * 16 + (lane & 15)] : 0.f;
    if (real && e + nwaves < EE)
      __builtin_prefetch(&ea[(size_t)(e + nwaves) * 16], 0, 1);  // global_prefetch

#pragma unroll
    for (int g = 0; g < 6; ++g) {
      float ep0, ep1;
      if (real) {
        ep0 = 0.f; ep1 = 0.f;
#pragma unroll
        for (int k = 0; k < 16; ++k) {
          const float ek = __shfl(eav, k, 32);
          ep0 += ek * WeLds[g * 1024 + k * 64 + c0];
          ep1 += ek * WeLds[g * 1024 + k * 64 + c1];
        }
      } else {
        ep0 = epmLds[g * 64 + c0];
        ep1 = epmLds[g * 64 + c1];
      }
      const float* plane = (g & 1) ? outh : outx;
      const int q = g >> 1;
      const float* sB = plane + (size_t)src * 384 + q * 128;        // Wl transform (x_j)
      const float* dB = plane + (size_t)dst * 384 + q * 128 + 64;   // Wr transform (x_i)
      const float m0 = lrelu(sB[c0] + dB[c0] + ep0);
      const float m1 = lrelu(sB[c1] + dB[c1] + ep1);
      const float a0 = wave_sum32(m0 * attLds[g * 64 + c0]);  // head 0
      const float a1 = wave_sum32(m1 * attLds[g * 64 + c1]);  // head 1
      if (lane == 0) {
        alphaB[(size_t)e * 12 + g * 2 + 0] = a0;
        alphaB[(size_t)e * 12 + g * 2 + 1] = a1;
        atomicMaxF(&amax[(size_t)dst * 12 + g * 2 + 0], a0);
        atomicMaxF(&amax[(size_t)dst * 12 + g * 2 + 1], a1);
      }
    }
  }
}

// ---------------------------------------------------------------------------
// K6: edge pass B — ex = exp(alpha - amax); accumulate Σex and Σex·x_j
//     (softmax normalization deferred to final node kernel)
// ---------------------------------------------------------------------------
__global__ void k_edge_b(const int* __restrict__ ei, const float* __restrict__ outx,
                         const float* __restrict__ outh, const float* __restrict__ alphaB,
                         const float* __restrict__ amax, float* __restrict__ denom,
                         float* __restrict__ outacc) {
  const int lane = threadIdx.x & 31;
  const int c0 = lane, c1 = lane + 32;
  const int wid = (blockIdx.x * blockDim.x + threadIdx.x) >> 5;
  const int nwaves = (gridDim.x * blockDim.x) >> 5;

  for (int e = wid; e < ETOT; e += nwaves) {
    int src, dst;
    if (e < EE) { src = ei[e]; dst = ei[EE + e]; }
    else        { src = e - EE; dst = src; }

#pragma unroll
    for (int g = 0; g < 6; ++g) {
      const float ex0 = expf(alphaB[(size_t)e * 12 + g * 2 + 0] -
                             amax[(size_t)dst * 12 + g * 2 + 0]);
      const float ex1 = expf(alphaB[(size_t)e * 12 + g * 2 + 1] -
                             amax[(size_t)dst * 12 + g * 2 + 1]);
      if (lane == 0) atomicAdd(&denom[(size_t)dst * 12 + g * 2 + 0], ex0);
      if (lane == 1) atomicAdd(&denom[(size_t)dst * 12 + g * 2 + 1], ex1);
      const float* plane = (g & 1) ? outh : outx;
      const float* sB = plane + (size_t)src * 384 + (g >> 1) * 128;
      float* acc = outacc + ((size_t)g * NN + dst) * 64;
      atomicAdd(&acc[c0], sB[c0] * ex0);
      atomicAdd(&acc[c1], sB[c1] * ex1);
    }
  }
}

// ---------------------------------------------------------------------------
// K7: per-node GRU combine + LayerNorm (one wave per node)
// ---------------------------------------------------------------------------
__global__ void k_final(const float* __restrict__ h_prev, const float* __restrict__ outacc,
                        const float* __restrict__ denom, AllPtrs P, float* __restrict__ out) {
  const int lane = threadIdx.x & 31;
  const int c0 = lane, c1 = lane + 32;
  const int wid = (blockIdx.x * blockDim.x + threadIdx.x) >> 5;
  const int nwaves = (gridDim.x * blockDim.x) >> 5;

  for (int n = wid; n < NN; n += nwaves) {
    float a0[6], a1[6];
#pragma unroll
    for (int g = 0; g < 6; ++g) {
      const float d0 = denom[(size_t)n * 12 + g * 2 + 0] + 1e-16f;
      const float d1 = denom[(size_t)n * 12 + g * 2 + 1] + 1e-16f;
      const float* acc = outacc + ((size_t)g * NN + n) * 64;
      a0[g] = acc[c0] / d0 + P.b[g][c0];
      a1[g] = acc[c1] / d1 + P.b[g][c1];
    }
    const float r0 = 1.f / (1.f + expf(-(a0[0] + a0[1])));
    const float r1 = 1.f / (1.f + expf(-(a1[0] + a1[1])));
    const float u0 = 1.f / (1.f + expf(-(a0[2] + a0[3])));
    const float u1 = 1.f / (1.f + expf(-(a1[2] + a1[3])));
    const float hh0 = tanhf(a0[4] + r0 * a0[5]);
    const float hh1 = tanhf(a1[4] + r1 * a1[5]);
    const float hp0 = h_prev[(size_t)n * 64 + c0];
    const float hp1 = h_prev[(size_t)n * 64 + c1];
    const float h0 = u0 * hp0 + (1.f - u0) * hh0;
    const float h1 = u1 * hp1 + (1.f - u1) * hh1;

    const float s1 = wave_sum32(h0 + h1);
    const float s2 = wave_sum32(h0 * h0 + h1 * h1);
    const float mu = s1 * (1.f / 64.f);
    const float var = s2 * (1.f / 64.f) - mu * mu;
    const float inv = rsqrtf(var + LN_EPS);
    out[(size_t)n * 64 + c0] = (h0 - mu) * inv * P.ln_g[c0] + P.ln_b[c0];
    out[(size_t)n * 64 + c1] = (h1 - mu) * inv * P.ln_g[c1] + P.ln_b[c1];
  }
}

// ---------------------------------------------------------------------------
extern "C" void kernel_launch(void* const* d_in, const int* in_sizes, int n_in,
                              void* d_out, int out_size, void* d_ws, size_t ws_size,
                              hipStream_t stream) {
  (void)in_sizes; (void)n_in; (void)out_size; (void)ws_size;
  const float* x_t    = (const float*)d_in[0];
  const float* h_prev = (const float*)d_in[1];
  const int*   ei     = (const int*)d_in[2];
  const float* ea     = (const float*)d_in[3];

  AllPtrs P;
  for (int g = 0; g < 6; ++g) {
    const int base = 4 + 7 * g;
    P.Wl[g]  = (const float*)d_in[base + 0];
    P.bl[g]  = (const float*)d_in[base + 1];
    P.Wr[g]  = (const float*)d_in[base + 2];
    P.br[g]  = (const float*)d_in[base + 3];
    P.We[g]  = (const float*)d_in[base + 4];
    P.att[g] = (const float*)d_in[base + 5];
    P.b[g]   = (const float*)d_in[base + 6];
  }
  P.ln_g = (const float*)d_in[46];
  P.ln_b = (const float*)d_in[47];

  float* W = (float*)d_ws;
  size_t off = 0;
  auto alloc = [&](size_t nelem) { float* p = W + off; off += nelem; return p; };
  float* out_x  = alloc((size_t)NN * 384);
  float* out_h  = alloc((size_t)NN * 384);
  float* outacc = alloc((size_t)6 * NN * 64);
  float* alphaB = alloc((size_t)ETOT * 12);
  float* amax   = alloc((size_t)NN * 12);
  float* denom  = alloc((size_t)NN * 12);
  float* Wx     = alloc(64 * 384);
  float* Wh     = alloc(64 * 384);
  float* bx     = alloc(384);
  float* bh     = alloc(384);
  float* eaSum  = alloc(16);
  float* epmean = alloc(384);

  k_init<<<4096, 256, 0, stream>>>(amax, denom, outacc, eaSum);
  k_ea_mean<<<4096, 256, 0, stream>>>(ea, eaSum);
  k_prep<<<1, 384, 0, stream>>>(P, eaSum, epmean);
  k_pack<<<((64 * 384 + 384) + 255) / 256, 256, 0, stream>>>(P, Wx, bx, Wh, bh);

  const int tiles = ((NN + 15) / 16) * 24;        // 75000 tiles
  const int gemmBlocks = (tiles + 7) / 8;         // 8 waves / 256-thread block
  k_gemm<<<gemmBlocks, 256, 0, stream>>>(x_t, Wx, bx, out_x, NN);
  k_gemm<<<gemmBlocks, 256, 0, stream>>>(h_prev, Wh, bh, out_h, NN);

  const int edgeBlocks = (ETOT + 7) / 8;          // 1 wave per edge
  k_edge_a<<<edgeBlocks, 256, 0, stream>>>(ei, ea, out_x, out_h, epmean, P, alphaB, amax);
  k_edge_b<<<edgeBlocks, 256, 0, stream>>>(ei, out_x, out_h, alphaB, amax, denom, outacc);

  k_final<<<(NN + 7) / 8, 256, 0, stream>>>(h_prev, outacc, denom, P, (float*)d_out);
}